// GraphGPSLayer_78383153152257
// MI455X (gfx1250) — compile-verified
//
#include <hip/hip_runtime.h>
#include <math.h>

// GraphGPS layer for MI455X (gfx1250), wave32 / WMMA / async-LDS staging.
//
// - All GEMM-shaped work (5 projections + both attention matmuls) runs on
//   V_WMMA_F32_16X16X4_F32 (exact fp32, matches the fp32 reference).
// - LDS tiles are staged with GLOBAL_LOAD_ASYNC_TO_LDS_B128 (ASYNCcnt) and
//   double-buffered: chunk i+1 streams into LDS while WMMAs consume chunk i.
//   Falls back to synchronous float4 staging if the builtin is unavailable.
// - Attention is flash-style: 64-row Q tiles per block, 64-col KV tiles
//   streamed through LDS with async prefetch, online softmax. The 512MB
//   score tensor is never materialized.
// - GCN edge aggregation uses hardware global_atomic_add_f32 (relaxed/agent);
//   xw (4MB) is resident in the 192MB L2 during the scatter.

#define N_NODES 4096
#define D_MODEL 256
#define N_HEADS 8
#define D_HEAD  32

typedef float v2f __attribute__((ext_vector_type(2)));
typedef float v8f __attribute__((ext_vector_type(8)));
typedef int   v4i __attribute__((ext_vector_type(4)));

#define GEMM_BIAS 1
#define GEMM_RELU 2
#define GEMM_RES  4

#if __has_builtin(__builtin_amdgcn_global_load_async_to_lds_b128)
#define HAS_ASYNC_LDS 1
#else
#define HAS_ASYNC_LDS 0
#endif

static __device__ __forceinline__ v8f wmma_f32_16x16x4(v2f a, v2f b, v8f c) {
  // 8 args: (neg_a, A, neg_b, B, c_mod, C, reuse_a, reuse_b)
  return __builtin_amdgcn_wmma_f32_16x16x4_f32(false, a, false, b, (short)0, c,
                                               false, false);
}

// 16B global -> LDS, ASYNCcnt-tracked (no VGPR round trip) when available.
static __device__ __forceinline__ void stage_b128(const float* g, float* l) {
#if HAS_ASYNC_LDS
  __builtin_amdgcn_global_load_async_to_lds_b128(
      (__attribute__((address_space(1))) v4i*)g,
      (__attribute__((address_space(3))) v4i*)l, 0, 0);
#else
  *(float4*)l = *(const float4*)g;
#endif
}

static __device__ __forceinline__ void async_wait_all() {
#if HAS_ASYNC_LDS
  asm volatile("s_wait_asynccnt 0" ::: "memory");
#endif
}

static __device__ __forceinline__ void atomic_add_f32(float* p, float v) {
  __hip_atomic_fetch_add(p, v, __ATOMIC_RELAXED, __HIP_MEMORY_SCOPE_AGENT);
}

// ---------------------------------------------------------------------------
// LayerNorm: one block (256 threads == D) per row.
// ---------------------------------------------------------------------------
__global__ __launch_bounds__(256) void ln_kernel(const float* __restrict__ x,
                                                 const float* __restrict__ g,
                                                 const float* __restrict__ b,
                                                 float* __restrict__ y) {
  const int row = blockIdx.x, tid = threadIdx.x;
  __shared__ float red[256];
  float v = x[(size_t)row * D_MODEL + tid];
  red[tid] = v;
  __syncthreads();
  for (int s = 128; s > 0; s >>= 1) {
    if (tid < s) red[tid] += red[tid + s];
    __syncthreads();
  }
  float mean = red[0] * (1.0f / D_MODEL);
  __syncthreads();
  float d = v - mean;
  red[tid] = d * d;
  __syncthreads();
  for (int s = 128; s > 0; s >>= 1) {
    if (tid < s) red[tid] += red[tid + s];
    __syncthreads();
  }
  float var = red[0] * (1.0f / D_MODEL);
  y[(size_t)row * D_MODEL + tid] = d * rsqrtf(var + 1e-5f) * g[tid] + b[tid];
}

// ---------------------------------------------------------------------------
// C[M,Nn] = A[M,K] @ W[Nn,K]^T (+bias)(+relu)(+residual)
// Block = 256 threads = 8 waves (4x2), block tile 64x64, K chunked by 64,
// double-buffered async LDS staging. Each wave: 16 rows x 32 cols.
// ---------------------------------------------------------------------------
__global__ __launch_bounds__(256) void wmma_gemm_kernel(
    const float* __restrict__ A, const float* __restrict__ W,
    const float* __restrict__ bias, const float* __restrict__ res,
    float* __restrict__ C, int M, int Nn, int K, int flags) {
  __shared__ __align__(16) float As[2][64][68];  // +4 pad: float4-aligned rows
  __shared__ __align__(16) float Ws[2][64][68];
  const int tid = threadIdx.x;
  const int lane = tid & 31, wave = tid >> 5;
  const int wm = wave >> 1, wn = wave & 1;
  const int lm = lane & 15, lg = lane >> 4;  // half-wave group (K pair select)
  const int m0 = blockIdx.y * 64, n0 = blockIdx.x * 64;

  auto stage = [&](int k0, int buf) {
    for (int t = tid; t < 1024; t += 256) {  // 64x64 floats as float4
      int r = t >> 4, c4 = (t & 15) << 2;
      stage_b128(&A[(size_t)(m0 + r) * K + k0 + c4], &As[buf][r][c4]);
      stage_b128(&W[(size_t)(n0 + r) * K + k0 + c4], &Ws[buf][r][c4]);
    }
  };

  v8f acc0 = {0.f, 0.f, 0.f, 0.f, 0.f, 0.f, 0.f, 0.f};
  v8f acc1 = {0.f, 0.f, 0.f, 0.f, 0.f, 0.f, 0.f, 0.f};

  const int nchunks = K >> 6;
  stage(0, 0);
  for (int ci = 0; ci < nchunks; ++ci) {
    async_wait_all();
    __syncthreads();  // buffer ci&1 ready; buffer (ci+1)&1 free (read at ci-1)
    if (ci + 1 < nchunks) stage((ci + 1) << 6, (ci + 1) & 1);
    const int buf = ci & 1;
#pragma unroll
    for (int kk = 0; kk < 64; kk += 4) {
      // f32 A frag 16x4: lanes 0-15 hold K={kk,kk+1}, lanes 16-31 K={kk+2,+3}
      int ks = kk + 2 * lg;
      v2f a;
      a.x = As[buf][16 * wm + lm][ks];
      a.y = As[buf][16 * wm + lm][ks + 1];
      v2f b0, b1;
      b0.x = Ws[buf][32 * wn + lm][ks];
      b0.y = Ws[buf][32 * wn + lm][ks + 1];
      b1.x = Ws[buf][32 * wn + 16 + lm][ks];
      b1.y = Ws[buf][32 * wn + 16 + lm][ks + 1];
      acc0 = wmma_f32_16x16x4(a, b0, acc0);
      acc1 = wmma_f32_16x16x4(a, b1, acc1);
    }
  }
  // C/D layout: VGPR i -> (M = i + 8*lg, N = lm)
#pragma unroll
  for (int i = 0; i < 8; ++i) {
    int r = m0 + 16 * wm + i + 8 * lg;
    int c0 = n0 + 32 * wn + lm;
    float x0 = acc0[i], x1 = acc1[i];
    if (flags & GEMM_BIAS) { x0 += bias[c0]; x1 += bias[c0 + 16]; }
    if (flags & GEMM_RELU) { x0 = fmaxf(x0, 0.f); x1 = fmaxf(x1, 0.f); }
    if (flags & GEMM_RES) {
      x0 += res[(size_t)r * Nn + c0];
      x1 += res[(size_t)r * Nn + c0 + 16];
    }
    C[(size_t)r * Nn + c0] = x0;
    C[(size_t)r * Nn + c0 + 16] = x1;
  }
}

// ---------------------------------------------------------------------------
// Flash attention, fp32 WMMA, async double-buffered KV tiles.
// Grid = (N/64, H). Block = 256 threads = 8 waves.
// ---------------------------------------------------------------------------
__global__ __launch_bounds__(256) void flash_attn_kernel(
    const float* __restrict__ qkv, float* __restrict__ o) {
  const int head = blockIdx.y;
  const int i0 = blockIdx.x * 64;
  const int tid = threadIdx.x;
  const int lane = tid & 31, wave = tid >> 5;
  const int wm = wave >> 1, wn = wave & 1;
  const int lm = lane & 15, lg = lane >> 4;

  __shared__ __align__(16) float Qs[64][36];
  __shared__ __align__(16) float Ks[2][64][36];
  __shared__ __align__(16) float Vs[2][64][36];
  __shared__ __align__(16) float Os[64][36];
  __shared__ __align__(16) float Ss[64][68];
  __shared__ float pmax[64][4], psum[64][4];
  __shared__ float mrow[64], lrow[64], arow[64];

  const float scale = 0.17677669529663687f;  // 1/sqrt(32)
  const size_t ld = 3 * D_MODEL;
  const size_t qoff = (size_t)head * D_HEAD;

  auto stageKV = [&](int j0, int buf) {
    for (int t = tid; t < 512; t += 256) {  // 64 rows x 8 float4
      int r = t >> 3, c4 = (t & 7) << 2;
      stage_b128(&qkv[(size_t)(j0 + r) * ld + D_MODEL + qoff + c4],
                 &Ks[buf][r][c4]);
      stage_b128(&qkv[(size_t)(j0 + r) * ld + 2 * D_MODEL + qoff + c4],
                 &Vs[buf][r][c4]);
    }
  };

  // stage Q (unscaled; scale folded into S epilogue), prefetch KV tile 0
  for (int t = tid; t < 512; t += 256) {
    int r = t >> 3, c4 = (t & 7) << 2;
    stage_b128(&qkv[(size_t)(i0 + r) * ld + qoff + c4], &Qs[r][c4]);
    *(float4*)&Os[r][c4] = make_float4(0.f, 0.f, 0.f, 0.f);
  }
  stageKV(0, 0);
  if (tid < 64) { mrow[tid] = -3.0e38f; lrow[tid] = 0.f; }

  const int ntiles = N_NODES / 64;
  for (int jt = 0; jt < ntiles; ++jt) {
    async_wait_all();
    __syncthreads();  // Q + KV[jt&1] ready; prior iteration fully consumed
    if (jt + 1 < ntiles) stageKV((jt + 1) * 64, (jt + 1) & 1);
    const int buf = jt & 1;

    // ---- S = Q @ K^T : 64x64, K-dim = 32 (scale applied in epilogue)
    v8f s0 = {0.f, 0.f, 0.f, 0.f, 0.f, 0.f, 0.f, 0.f};
    v8f s1 = {0.f, 0.f, 0.f, 0.f, 0.f, 0.f, 0.f, 0.f};
#pragma unroll
    for (int kk = 0; kk < 32; kk += 4) {
      int ks = kk + 2 * lg;
      v2f a;
      a.x = Qs[16 * wm + lm][ks];
      a.y = Qs[16 * wm + lm][ks + 1];
      v2f b0, b1;
      b0.x = Ks[buf][32 * wn + lm][ks];
      b0.y = Ks[buf][32 * wn + lm][ks + 1];
      b1.x = Ks[buf][32 * wn + 16 + lm][ks];
      b1.y = Ks[buf][32 * wn + 16 + lm][ks + 1];
      s0 = wmma_f32_16x16x4(a, b0, s0);
      s1 = wmma_f32_16x16x4(a, b1, s1);
    }
#pragma unroll
    for (int i = 0; i < 8; ++i) {
      int r = 16 * wm + i + 8 * lg;
      Ss[r][32 * wn + lm] = s0[i] * scale;
      Ss[r][32 * wn + 16 + lm] = s1[i] * scale;
    }
    __syncthreads();

    // ---- online softmax: 4 threads per row (16 cols each)
    const int r = tid >> 2, q = tid & 3, c0 = q << 4;
    float mj = -3.0e38f;
#pragma unroll
    for (int c = 0; c < 16; ++c) mj = fmaxf(mj, Ss[r][c0 + c]);
    pmax[r][q] = mj;
    __syncthreads();
    float mold = mrow[r];
    float mnew = fmaxf(mold, fmaxf(fmaxf(pmax[r][0], pmax[r][1]),
                                   fmaxf(pmax[r][2], pmax[r][3])));
    float sum = 0.f;
#pragma unroll
    for (int c = 0; c < 16; ++c) {
      float p = __expf(Ss[r][c0 + c] - mnew);
      Ss[r][c0 + c] = p;
      sum += p;
    }
    psum[r][q] = sum;
    __syncthreads();
    if (q == 0) {
      float al = __expf(mold - mnew);
      arow[r] = al;
      lrow[r] = lrow[r] * al + psum[r][0] + psum[r][1] + psum[r][2] + psum[r][3];
      mrow[r] = mnew;
    }
    __syncthreads();

    // ---- O = O*alpha + P @ V : 64x32, K-dim = 64 (one 16x16 tile per wave)
    v8f po = {0.f, 0.f, 0.f, 0.f, 0.f, 0.f, 0.f, 0.f};
#pragma unroll
    for (int kk = 0; kk < 64; kk += 4) {
      int ks = kk + 2 * lg;
      v2f a;
      a.x = Ss[16 * wm + lm][ks];
      a.y = Ss[16 * wm + lm][ks + 1];
      v2f b;
      b.x = Vs[buf][ks][16 * wn + lm];
      b.y = Vs[buf][ks + 1][16 * wn + lm];
      po = wmma_f32_16x16x4(a, b, po);
    }
#pragma unroll
    for (int i = 0; i < 8; ++i) {
      int rr = 16 * wm + i + 8 * lg;
      int cc = 16 * wn + lm;
      Os[rr][cc] = Os[rr][cc] * arow[rr] + po[i];  // rescale fused here
    }
  }
  __syncthreads();

  for (int t = tid; t < 2048; t += 256) {
    int r = t >> 5, c = t & 31;
    o[(size_t)(i0 + r) * D_MODEL + head * D_HEAD + c] = Os[r][c] / lrow[r];
  }
}

// ---------------------------------------------------------------------------
// GCN sparse aggregation helpers
// ---------------------------------------------------------------------------
__global__ __launch_bounds__(256) void deg_init_kernel(float* __restrict__ deg) {
  deg[blockIdx.x * 256 + threadIdx.x] = 1.0f;  // self-loop
}

__global__ __launch_bounds__(256) void deg_accum_kernel(
    const long long* __restrict__ dst, float* __restrict__ deg, int E) {
  int e = blockIdx.x * 256 + threadIdx.x;
  if (e < E) atomic_add_f32(&deg[(int)dst[e]], 1.0f);
}

__global__ __launch_bounds__(256) void dinv_kernel(float* __restrict__ deg) {
  int n = blockIdx.x * 256 + threadIdx.x;
  deg[n] = rsqrtf(deg[n]);  // deg >= 1 always (self-loop)
}

// hloc = xw * dinv[n]^2  (self-loop contribution)
__global__ __launch_bounds__(256) void selfloop_kernel(
    const float* __restrict__ xw, const float* __restrict__ dinv,
    float* __restrict__ hloc) {
  size_t i = (size_t)blockIdx.x * 256 + threadIdx.x;
  int n = (int)(i >> 8);  // D == 256
  float di = dinv[n];
  hloc[i] = xw[i] * di * di;
}

// One block per edge; 256 lanes = 256 features. xw (4MB) is L2-resident.
__global__ __launch_bounds__(256) void scatter_kernel(
    const float* __restrict__ xw, const long long* __restrict__ ei,
    const float* __restrict__ dinv, float* __restrict__ hloc, int E) {
  int e = blockIdx.x;
  int s = (int)ei[e];
  int d = (int)ei[(size_t)E + e];
  float coef = dinv[s] * dinv[d];
  atomic_add_f32(&hloc[(size_t)d * D_MODEL + threadIdx.x],
                 xw[(size_t)s * D_MODEL + threadIdx.x] * coef);
}

// h1 = h + hloc + gcn_b
__global__ __launch_bounds__(256) void add_bias_res_kernel(
    const float* __restrict__ h, const float* __restrict__ hloc,
    const float* __restrict__ b, float* __restrict__ h1) {
  size_t i = (size_t)blockIdx.x * 256 + threadIdx.x;
  h1[i] = h[i] + hloc[i] + b[threadIdx.x];  // blockDim == D == 256
}

// ---------------------------------------------------------------------------
extern "C" void kernel_launch(void* const* d_in, const int* in_sizes, int n_in,
                              void* d_out, int out_size, void* d_ws,
                              size_t ws_size, hipStream_t stream) {
  const float* h = (const float*)d_in[0];
  const long long* ei = (const long long*)d_in[1];  // int64 [2,E]
  const float* gcn_w = (const float*)d_in[2];
  const float* gcn_b = (const float*)d_in[3];
  const float* in_proj_w = (const float*)d_in[4];
  const float* in_proj_b = (const float*)d_in[5];
  const float* out_proj_w = (const float*)d_in[6];
  const float* out_proj_b = (const float*)d_in[7];
  const float* ln1_g = (const float*)d_in[8];
  const float* ln1_b = (const float*)d_in[9];
  const float* ln2_g = (const float*)d_in[10];
  const float* ln2_b = (const float*)d_in[11];
  const float* ln3_g = (const float*)d_in[12];
  const float* ln3_b = (const float*)d_in[13];
  const float* ffn_w1 = (const float*)d_in[14];
  const float* ffn_b1 = (const float*)d_in[15];
  const float* ffn_w2 = (const float*)d_in[16];
  const float* ffn_b2 = (const float*)d_in[17];
  const int E = in_sizes[1] / 2;

  const size_t ND = (size_t)N_NODES * D_MODEL;
  float* ws = (float*)d_ws;  // ~46MB used
  float* hn = ws;            // ND
  float* xw = hn + ND;       // ND
  float* hloc = xw + ND;     // ND
  float* h1 = hloc + ND;     // ND
  float* qkv = h1 + ND;      // 3*ND
  float* ob = qkv + 3 * ND;  // ND
  float* h2 = ob + ND;       // ND
  float* mid = h2 + ND;      // 2*ND
  float* dinv = mid + 2 * ND;  // N
  float* out = (float*)d_out;

  dim3 b256(256);

  // --- GCN block ---
  ln_kernel<<<N_NODES, b256, 0, stream>>>(h, ln1_g, ln1_b, hn);
  wmma_gemm_kernel<<<dim3(D_MODEL / 64, N_NODES / 64), b256, 0, stream>>>(
      hn, gcn_w, nullptr, nullptr, xw, N_NODES, D_MODEL, D_MODEL, 0);
  deg_init_kernel<<<N_NODES / 256, b256, 0, stream>>>(dinv);
  deg_accum_kernel<<<(E + 255) / 256, b256, 0, stream>>>(ei + E, dinv, E);
  dinv_kernel<<<N_NODES / 256, b256, 0, stream>>>(dinv);
  selfloop_kernel<<<(unsigned)(ND / 256), b256, 0, stream>>>(xw, dinv, hloc);
  scatter_kernel<<<E, b256, 0, stream>>>(xw, ei, dinv, hloc, E);
  add_bias_res_kernel<<<(unsigned)(ND / 256), b256, 0, stream>>>(h, hloc, gcn_b,
                                                                 h1);

  // --- global attention block ---
  ln_kernel<<<N_NODES, b256, 0, stream>>>(h1, ln2_g, ln2_b, hn);
  wmma_gemm_kernel<<<dim3((3 * D_MODEL) / 64, N_NODES / 64), b256, 0, stream>>>(
      hn, in_proj_w, in_proj_b, nullptr, qkv, N_NODES, 3 * D_MODEL, D_MODEL,
      GEMM_BIAS);
  flash_attn_kernel<<<dim3(N_NODES / 64, N_HEADS), b256, 0, stream>>>(qkv, ob);
  wmma_gemm_kernel<<<dim3(D_MODEL / 64, N_NODES / 64), b256, 0, stream>>>(
      ob, out_proj_w, out_proj_b, h1, h2, N_NODES, D_MODEL, D_MODEL,
      GEMM_BIAS | GEMM_RES);

  // --- FFN block ---
  ln_kernel<<<N_NODES, b256, 0, stream>>>(h2, ln3_g, ln3_b, hn);
  wmma_gemm_kernel<<<dim3((2 * D_MODEL) / 64, N_NODES / 64), b256, 0, stream>>>(
      hn, ffn_w1, ffn_b1, nullptr, mid, N_NODES, 2 * D_MODEL, D_MODEL,
      GEMM_BIAS | GEMM_RELU);
  wmma_gemm_kernel<<<dim3(D_MODEL / 64, N_NODES / 64), b256, 0, stream>>>(
      mid, ffn_w2, ffn_b2, h2, out, N_NODES, D_MODEL, 2 * D_MODEL,
      GEMM_BIAS | GEMM_RES);
}